// GATEncoder_35983236005888
// MI455X (gfx1250) — compile-verified
//
#include <hip/hip_runtime.h>
#include <math.h>

typedef __attribute__((ext_vector_type(2))) float v2f;
typedef __attribute__((ext_vector_type(8))) float v8f;

#define LRELU_SLOPE 0.2f
#define TPB 256

// ---- float atomic max via int/uint ordering trick (handles mixed signs) ----
__device__ __forceinline__ void atomicMaxF32(float* addr, float v) {
  if (v >= 0.0f) atomicMax((int*)addr, __float_as_int(v));
  else           atomicMin((unsigned int*)addr, __float_as_uint(v));
}

// ---------------- zero fill ----------------
__global__ void k_zero(float* __restrict__ p, long long n) {
  long long i = (long long)blockIdx.x * blockDim.x + threadIdx.x;
  long long stride = (long long)gridDim.x * blockDim.x;
  for (; i < n; i += stride) p[i] = 0.0f;
}

// ---------------- fp32 WMMA GEMM: C[M,N] = A[M,K] @ B[K,N] ----------------
// One wave32 per 16x16 output tile; K stepped by 4 via V_WMMA_F32_16X16X4_F32.
// A frag (16x4, 32-bit): lanes 0-15 -> K=0(v0),K=1(v1); lanes 16-31 -> K=2,K=3.
// B frag (4x16): same K split across lane halves, N = lane&15.
// C/D (16x16 f32): VGPR r -> M = r + 8*(lane>>4), N = lane&15.
__global__ void k_gemm_wmma(const float* __restrict__ A, const float* __restrict__ B,
                            float* __restrict__ C, int M, int N, int K) {
  int wave = (int)((blockIdx.x * blockDim.x + threadIdx.x) >> 5);
  int lane = threadIdx.x & 31;
  int ntn  = N >> 4;
  int tm = wave / ntn;
  int tn = wave - tm * ntn;
  if (tm * 16 >= M) return;  // wave-uniform: EXEC stays all-ones for WMMA

  int mrow = tm * 16 + (lane & 15);
  int ncol = tn * 16 + (lane & 15);
  int kh   = (lane >> 4) << 1;  // 0 or 2
  const float* arow = A + (long long)mrow * K;

  v8f c = {};
  for (int k = 0; k < K; k += 4) {
    v2f a, b;
    a.x = arow[k + kh];
    a.y = arow[k + kh + 1];
    b.x = B[(long long)(k + kh) * N + ncol];
    b.y = B[(long long)(k + kh + 1) * N + ncol];
    c = __builtin_amdgcn_wmma_f32_16x16x4_f32(false, a, false, b, (short)0, c,
                                              false, false);
  }
  int row0 = tm * 16 + ((lane >> 4) << 3);
  int col  = tn * 16 + (lane & 15);
#pragma unroll
  for (int r = 0; r < 8; ++r)
    C[(long long)(row0 + r) * N + col] = c[r];
}

// ---- per-node attention halves + init of segment max / segment sum ----
__global__ void k_alpha(const float* __restrict__ h, const float* __restrict__ avs,
                        const float* __restrict__ avd, float* __restrict__ as,
                        float* __restrict__ ad, float* __restrict__ mmax,
                        float* __restrict__ ssum, int N, int D) {
  int wave = (int)((blockIdx.x * blockDim.x + threadIdx.x) >> 5);
  int lane = threadIdx.x & 31;
  if (wave >= N) return;
  const float* row = h + (long long)wave * D;
  float s1 = 0.0f, s2 = 0.0f;
  for (int f = lane; f < D; f += 32) {
    float v = row[f];
    s1 += v * avs[f];
    s2 += v * avd[f];
  }
#pragma unroll
  for (int off = 16; off; off >>= 1) {
    s1 += __shfl_xor(s1, off);
    s2 += __shfl_xor(s2, off);
  }
  if (lane == 0) {
    as[wave] = s1;
    ad[wave] = s2;
    mmax[wave] = -INFINITY;
    ssum[wave] = 0.0f;
  }
}

__device__ __forceinline__ void edge_ids(long long t, long long E,
                                         const int* __restrict__ src,
                                         const int* __restrict__ dst,
                                         int& s, int& d) {
  if (t < E) { s = src[t]; d = dst[t]; }
  else       { int i = (int)(t - E); s = i; d = i; }  // self-loops
}

// ---- pass 1: segment max of leaky_relu logits ----
__global__ void k_edge_max(const int* __restrict__ src, const int* __restrict__ dst,
                           const float* __restrict__ as, const float* __restrict__ ad,
                           float* __restrict__ mmax, long long E, long long NE) {
  long long t = (long long)blockIdx.x * blockDim.x + threadIdx.x;
  if (t >= NE) return;
  int s, d; edge_ids(t, E, src, dst, s, d);
  float e = as[s] + ad[d];
  e = (e > 0.0f) ? e : LRELU_SLOPE * e;
  atomicMaxF32(&mmax[d], e);
}

// ---- pass 2: segment sum of exp(e - max) ----
__global__ void k_edge_sum(const int* __restrict__ src, const int* __restrict__ dst,
                           const float* __restrict__ as, const float* __restrict__ ad,
                           const float* __restrict__ mmax, float* __restrict__ ssum,
                           long long E, long long NE) {
  long long t = (long long)blockIdx.x * blockDim.x + threadIdx.x;
  if (t >= NE) return;
  int s, d; edge_ids(t, E, src, dst, s, d);
  float e = as[s] + ad[d];
  e = (e > 0.0f) ? e : LRELU_SLOPE * e;
  atomicAdd(&ssum[d], __expf(e - mmax[d]));
}

// ---- pass 3: weighted scatter-add; one wave per edge, D = 128 (float4/lane) ----
__global__ void k_edge_agg128(const float* __restrict__ h, const int* __restrict__ src,
                              const int* __restrict__ dst, const float* __restrict__ as,
                              const float* __restrict__ ad, const float* __restrict__ mmax,
                              const float* __restrict__ ssum, float* __restrict__ out,
                              long long E, long long NE) {
  long long wv = ((long long)blockIdx.x * blockDim.x + threadIdx.x) >> 5;
  int lane = threadIdx.x & 31;
  if (wv >= NE) return;
  int s, d; edge_ids(wv, E, src, dst, s, d);
  float e = as[s] + ad[d];
  e = (e > 0.0f) ? e : LRELU_SLOPE * e;
  float w = __expf(e - mmax[d]) / (ssum[d] + 1e-16f);
  float4 v = ((const float4*)(h + (long long)s * 128))[lane];
  float* op = out + (long long)d * 128 + lane * 4;
  atomicAdd(op + 0, w * v.x);
  atomicAdd(op + 1, w * v.y);
  atomicAdd(op + 2, w * v.z);
  atomicAdd(op + 3, w * v.w);
}

// ---- pass 3 variant for D = 64 (float2/lane) ----
__global__ void k_edge_agg64(const float* __restrict__ h, const int* __restrict__ src,
                             const int* __restrict__ dst, const float* __restrict__ as,
                             const float* __restrict__ ad, const float* __restrict__ mmax,
                             const float* __restrict__ ssum, float* __restrict__ out,
                             long long E, long long NE) {
  long long wv = ((long long)blockIdx.x * blockDim.x + threadIdx.x) >> 5;
  int lane = threadIdx.x & 31;
  if (wv >= NE) return;
  int s, d; edge_ids(wv, E, src, dst, s, d);
  float e = as[s] + ad[d];
  e = (e > 0.0f) ? e : LRELU_SLOPE * e;
  float w = __expf(e - mmax[d]) / (ssum[d] + 1e-16f);
  float2 v = ((const float2*)(h + (long long)s * 64))[lane];
  float* op = out + (long long)d * 64 + lane * 2;
  atomicAdd(op + 0, w * v.x);
  atomicAdd(op + 1, w * v.y);
}

// ---- bias + ReLU, in-place ----
__global__ void k_bias_relu(float* __restrict__ p, const float* __restrict__ b,
                            long long n, int D) {
  long long i = (long long)blockIdx.x * blockDim.x + threadIdx.x;
  long long stride = (long long)gridDim.x * blockDim.x;
  for (; i < n; i += stride) {
    float v = p[i] + b[(int)(i % D)];
    p[i] = v > 0.0f ? v : 0.0f;
  }
}

// ---- layer-2 epilogue: bias + column mean, block-local LDS reduce then atomics ----
// Requires blockDim (256) % D == 0; grid stride is then a multiple of D so each
// thread owns one fixed column.
__global__ void k_bias_mean(const float* __restrict__ agg, const float* __restrict__ b,
                            float* __restrict__ out, long long total, int D, float invN) {
  __shared__ float sh[TPB];
  long long i = (long long)blockIdx.x * blockDim.x + threadIdx.x;
  long long stride = (long long)gridDim.x * blockDim.x;
  int col = (int)(i % D);
  float acc = 0.0f;
  for (; i < total; i += stride) acc += agg[i] + b[col];
  sh[threadIdx.x] = acc;
  __syncthreads();
  if ((int)threadIdx.x < D) {
    float s = 0.0f;
    for (int t = threadIdx.x; t < (int)blockDim.x; t += D) s += sh[t];
    atomicAdd(&out[threadIdx.x], s * invN);
  }
}

extern "C" void kernel_launch(void* const* d_in, const int* in_sizes, int n_in,
                              void* d_out, int out_size, void* d_ws, size_t ws_size,
                              hipStream_t stream) {
  (void)n_in; (void)ws_size;
  const float* x   = (const float*)d_in[0];
  const int*   ei  = (const int*)d_in[1];
  const float* W1  = (const float*)d_in[2];
  const float* a1s = (const float*)d_in[3];
  const float* a1d = (const float*)d_in[4];
  const float* b1  = (const float*)d_in[5];
  const float* W2  = (const float*)d_in[6];
  const float* a2s = (const float*)d_in[7];
  const float* a2d = (const float*)d_in[8];
  const float* b2  = (const float*)d_in[9];
  float* out = (float*)d_out;

  const int DH = in_sizes[3];                     // 128
  const int DI = in_sizes[2] / DH;                // 128
  const int DO = in_sizes[7];                     // 64
  const long long N = (long long)in_sizes[0] / DI;
  const long long E = (long long)in_sizes[1] / 2; // edge_index is [2, E] flat
  const int* src = ei;
  const int* dst = ei + E;

  // ---- carve workspace (256B aligned chunks) ----
  char* w = (char*)d_ws;
  auto alloc = [&](size_t bytes) -> float* {
    float* p = (float*)w;
    w += (bytes + 255) & ~(size_t)255;
    return p;
  };
  float* h1   = alloc((size_t)N * DH * 4);
  float* agg1 = alloc((size_t)N * DH * 4);
  float* h2   = alloc((size_t)N * DO * 4);
  float* agg2 = alloc((size_t)N * DO * 4);
  float* as   = alloc((size_t)N * 4);
  float* ad   = alloc((size_t)N * 4);
  float* mmax = alloc((size_t)N * 4);
  float* ssum = alloc((size_t)N * 4);

  const long long NE = E + N;                     // edges + self-loops
  const int eblocks   = (int)((NE + TPB - 1) / TPB);
  const int aggblocks = (int)((NE * 32 + TPB - 1) / TPB);
  const int ablocks   = (int)((N * 32 + TPB - 1) / TPB);

  // ---- init accumulators (every call: buffers are poisoned / reused) ----
  k_zero<<<2048, TPB, 0, stream>>>(agg1, N * DH);
  k_zero<<<1024, TPB, 0, stream>>>(agg2, N * DO);
  k_zero<<<1, 64, 0, stream>>>(out, out_size);

  // ================= layer 1 =================
  {
    int waves  = (int)((N / 16) * (DH / 16));
    int blocks = (waves * 32 + TPB - 1) / TPB;
    k_gemm_wmma<<<blocks, TPB, 0, stream>>>(x, W1, h1, (int)N, DH, DI);
    k_alpha<<<ablocks, TPB, 0, stream>>>(h1, a1s, a1d, as, ad, mmax, ssum, (int)N, DH);
    k_edge_max<<<eblocks, TPB, 0, stream>>>(src, dst, as, ad, mmax, E, NE);
    k_edge_sum<<<eblocks, TPB, 0, stream>>>(src, dst, as, ad, mmax, ssum, E, NE);
    k_edge_agg128<<<aggblocks, TPB, 0, stream>>>(h1, src, dst, as, ad, mmax, ssum,
                                                 agg1, E, NE);
    k_bias_relu<<<2048, TPB, 0, stream>>>(agg1, b1, N * DH, DH);
  }

  // ================= layer 2 =================
  {
    int waves  = (int)((N / 16) * (DO / 16));
    int blocks = (waves * 32 + TPB - 1) / TPB;
    k_gemm_wmma<<<blocks, TPB, 0, stream>>>(agg1, W2, h2, (int)N, DO, DH);
    k_alpha<<<ablocks, TPB, 0, stream>>>(h2, a2s, a2d, as, ad, mmax, ssum, (int)N, DO);
    k_edge_max<<<eblocks, TPB, 0, stream>>>(src, dst, as, ad, mmax, E, NE);
    k_edge_sum<<<eblocks, TPB, 0, stream>>>(src, dst, as, ad, mmax, ssum, E, NE);
    k_edge_agg64<<<aggblocks, TPB, 0, stream>>>(h2, src, dst, as, ad, mmax, ssum,
                                                agg2, E, NE);
    k_bias_mean<<<1024, TPB, 0, stream>>>(agg2, b2, out, N * DO, DO, 1.0f / (float)N);
  }
}